// PolarProjectionDepth_76484777607406
// MI455X (gfx1250) — compile-verified
//
#include <hip/hip_runtime.h>

// Problem constants (fixed by the reference's setup_inputs)
#define BB 8
#define DD 128
#define HH 64
#define WW 256
#define SS 32
#define ZZ 64
#define PAD 2   // row pad (floats): keeps 8B alignment for ds_load_b64, spreads banks

typedef __attribute__((ext_vector_type(2))) float v2f;
typedef __attribute__((ext_vector_type(8))) float v8f;

// CDNA5 async global->LDS copies (ASYNCcnt path), per CDNA5 ISA.
// VDST = LDS byte address (low 32 bits of the generic shared pointer), VADDR = 64-bit global.
__device__ __forceinline__ void async_g2l_b32(unsigned ldsoff, const float* g) {
    asm volatile("global_load_async_to_lds_b32 %0, %1, off"
                 :: "v"(ldsoff), "v"((unsigned long long)(size_t)g)
                 : "memory");
}
__device__ __forceinline__ void async_g2l_b128(unsigned ldsoff, const float* g) {
    asm volatile("global_load_async_to_lds_b128 %0, %1, off"
                 :: "v"(ldsoff), "v"((unsigned long long)(size_t)g)
                 : "memory");
}
__device__ __forceinline__ void wait_async0() {
    asm volatile("s_wait_asynccnt 0x0" ::: "memory");
}

__global__ __launch_bounds__(256) void
PolarProjectionDepth_76484777607406_kernel(const float* __restrict__ image,
                                           const float* __restrict__ pscales,
                                           float* __restrict__ out) {
    const int w   = blockIdx.x;   // 0..255
    const int b   = blockIdx.y;   // 0..7
    const int tid = threadIdx.x;  // 0..255

    __shared__ float ps[HH][SS];            //  8 KB: pixel_scales[b,:,w,:]
    __shared__ float prob_t[ZZ][HH + PAD];  // ~17 KB: softmax probs TRANSPOSED (z,h)
    __shared__ float img[DD][HH + PAD];     // ~34 KB: image[b,:, :, w]
    __shared__ float red[4][ZZ];            //  1 KB: per-column partial reductions

    // ---------- Phase A: async global -> LDS staging (ASYNCcnt) ----------
    // pixel_scales[b,h,w,s]: ((b*H+h)*W + w)*S + s ; each h-row is 32 floats = 128B contiguous
    #pragma unroll
    for (int it = 0; it < (HH * SS) / (256 * 4); ++it) {   // 2 iters, b128 chunks
        int c  = tid + it * 256;
        int h  = c >> 3;
        int sg = (c & 7) * 4;
        async_g2l_b128((unsigned)(size_t)&ps[h][sg],
                       &pscales[(((size_t)b * HH + h) * WW + w) * SS + sg]);
    }
    // image[b,d,h,w]: stride-W scatter for fixed w; lines are L2-resident across w-blocks
    #pragma unroll 4
    for (int it = 0; it < (DD * HH) / 256; ++it) {         // 32 iters, b32 transfers
        int i = tid + it * 256;
        int d = i >> 6, h = i & (HH - 1);
        async_g2l_b32((unsigned)(size_t)&img[d][h],
                      &image[(((size_t)b * DD + d) * HH + h) * WW + w]);
    }
    wait_async0();
    __syncthreads();

    // ---------- Phase B: grid-sample scores + softmax over H (per z column) ----------
    const int z    = tid & (ZZ - 1);
    const int part = tid >> 6;   // owns h in [16*part, 16*part+16)

    // depth_steps[::-1][z] = 32 - 0.5*z ; pix = (log2(144.4/depth) - 2) * 31/7
    float depth = 32.0f - 0.5f * (float)z;
    float pix   = (__log2f(144.4f / depth) - 2.0f) * (31.0f / 7.0f);
    int   i0    = (int)floorf(pix);
    int   i1    = i0 + 1;
    float w1    = pix - (float)i0;
    float w0f   = 1.0f - w1;
    float m0    = (i0 >= 0 && i0 <= SS - 1) ? 1.0f : 0.0f;
    float m1    = (i1 >= 0 && i1 <= SS - 1) ? 1.0f : 0.0f;
    int   i0c   = min(max(i0, 0), SS - 1);
    int   i1c   = min(max(i1, 0), SS - 1);
    float g0 = w0f * m0, g1 = w1 * m1;

    float sc[16];
    float lmax = -3.402823466e38f;
    #pragma unroll
    for (int j = 0; j < 16; ++j) {
        int h = part * 16 + j;
        float v = ps[h][i0c] * g0 + ps[h][i1c] * g1;
        sc[j] = v;
        lmax = fmaxf(lmax, v);
    }
    red[part][z] = lmax;
    __syncthreads();
    float cmax = fmaxf(fmaxf(red[0][z], red[1][z]), fmaxf(red[2][z], red[3][z]));
    __syncthreads();  // all reads of red done before reuse

    float lsum = 0.0f;
    #pragma unroll
    for (int j = 0; j < 16; ++j) {
        float e = __expf(sc[j] - cmax);
        prob_t[z][part * 16 + j] = e;   // transposed store: row = z, col = h
        lsum += e;
    }
    red[part][z] = lsum;
    __syncthreads();
    float inv = 1.0f / (red[0][z] + red[1][z] + red[2][z] + red[3][z]);
    #pragma unroll
    for (int j = 0; j < 16; ++j) {
        prob_t[z][part * 16 + j] *= inv;
    }
    __syncthreads();

    // ---------- Phase C: WMMA GEMM  out[d,z] = sum_h img[d,h] * prob_t[z,h] ----------
    // wave i (of 8) owns d-rows [16i, 16i+16); 4 N-tiles of 16 z each.
    // A 16x4 f32 layout: lanes 0-15 (M=lane): K0/K1 ; lanes 16-31 (M=lane-16): K2/K3
    // B 4x16  f32 layout: lanes 0-15 (N=lane): K0/K1 ; lanes 16-31 (N=lane-16): K2/K3
    const int wave  = tid >> 5;
    const int lane  = tid & 31;
    const int l15   = lane & 15;
    const int khalf = (lane >> 4) * 2;
    const int mrow  = wave * 16 + l15;

    v8f acc[4] = {};

    for (int k = 0; k < HH; k += 4) {
        v2f a = *(const v2f*)&img[mrow][k + khalf];          // ds_load_b64, contiguous
        #pragma unroll
        for (int n = 0; n < 4; ++n) {
            v2f bf = *(const v2f*)&prob_t[n * 16 + l15][k + khalf];  // ds_load_b64
            acc[n] = __builtin_amdgcn_wmma_f32_16x16x4_f32(
                /*neg_a=*/false, a, /*neg_b=*/false, bf,
                /*c_mod=*/(short)0, acc[n],
                /*reuse_a=*/false, /*reuse_b=*/false);
        }
    }

    // ---------- Phase D: scatter C tiles to out[b,d,z,w] ----------
    // C/D layout: VGPR j, lanes 0-15 -> (M=j, N=lane); lanes 16-31 -> (M=j+8, N=lane-16)
    const int dbase = wave * 16 + 8 * (lane >> 4);
    #pragma unroll
    for (int n = 0; n < 4; ++n) {
        int zz = n * 16 + l15;
        #pragma unroll
        for (int j = 0; j < 8; ++j) {
            int d = dbase + j;
            out[(((size_t)b * DD + d) * ZZ + zz) * WW + w] = acc[n][j];
        }
    }
}

extern "C" void kernel_launch(void* const* d_in, const int* in_sizes, int n_in,
                              void* d_out, int out_size, void* d_ws, size_t ws_size,
                              hipStream_t stream) {
    (void)in_sizes; (void)n_in; (void)out_size; (void)d_ws; (void)ws_size;
    const float* image   = (const float*)d_in[0];  // (B, D, H, W) f32
    const float* pscales = (const float*)d_in[1];  // (B, H, W, S) f32
    float* out = (float*)d_out;                    // (B, D, Z, W) f32

    dim3 grid(WW, BB, 1);   // one workgroup per (w, b)
    dim3 block(256, 1, 1);  // 8 wave32 waves
    PolarProjectionDepth_76484777607406_kernel<<<grid, block, 0, stream>>>(image, pscales, out);
}